// BPCAPooling_43868795961630
// MI455X (gfx1250) — compile-verified
//
#include <hip/hip_runtime.h>
#include <math.h>

// Problem geometry (fixed by the reference)
#define NBATCH 32
#define PH 128
#define PW 128
#define PC 64
#define NROWS (PH * PW * PC / 4)      // 262144 float4 "patch rows" per batch
#define IN_PER_B (NROWS * 4)          // 1048576 floats per batch
#define OUT_PER_B NROWS               // 262144 output floats per batch

// Reduction kernel config
#define BLKA 256
#define ROWS_PER_THREAD 8
#define CHUNKS (NROWS / (BLKA * ROWS_PER_THREAD))   // 128 blocks per batch

typedef __attribute__((ext_vector_type(2))) float v2f;
typedef __attribute__((ext_vector_type(8))) float v8f;

#if defined(__has_builtin)
#if __has_builtin(__builtin_amdgcn_wmma_f32_16x16x4_f32)
#define HAS_WMMA_F32_4 1
#endif
#endif

// ---------------------------------------------------------------------------
// Kernel 1: fused 14-way reduction (S0..S3 and 10 unique second moments).
// One pass over the 128 MB input. float4 loads, deterministic LDS tree.
// ---------------------------------------------------------------------------
__global__ __launch_bounds__(BLKA) void bpca_reduce(const float* __restrict__ in,
                                                    float* __restrict__ partials) {
  const int batch = blockIdx.y;
  const int chunk = blockIdx.x;
  const int tid = threadIdx.x;

  const float4* rows = (const float4*)(in + (size_t)batch * IN_PER_B);

  float a[14];
#pragma unroll
  for (int t = 0; t < 14; ++t) a[t] = 0.0f;

  int r = chunk * BLKA + tid;
#pragma unroll
  for (int i = 0; i < ROWS_PER_THREAD; ++i, r += BLKA * CHUNKS) {
    float4 p = rows[r];
    a[0] += p.x; a[1] += p.y; a[2] += p.z; a[3] += p.w;
    a[4]  = fmaf(p.x, p.x, a[4]);
    a[5]  = fmaf(p.x, p.y, a[5]);
    a[6]  = fmaf(p.x, p.z, a[6]);
    a[7]  = fmaf(p.x, p.w, a[7]);
    a[8]  = fmaf(p.y, p.y, a[8]);
    a[9]  = fmaf(p.y, p.z, a[9]);
    a[10] = fmaf(p.y, p.w, a[10]);
    a[11] = fmaf(p.z, p.z, a[11]);
    a[12] = fmaf(p.z, p.w, a[12]);
    a[13] = fmaf(p.w, p.w, a[13]);
  }

  __shared__ float red[14][BLKA];
#pragma unroll
  for (int t = 0; t < 14; ++t) red[t][tid] = a[t];
  __syncthreads();

  for (int off = BLKA / 2; off > 0; off >>= 1) {
    if (tid < off) {
#pragma unroll
      for (int t = 0; t < 14; ++t) red[t][tid] += red[t][tid + off];
    }
    __syncthreads();
  }

  if (tid == 0) {
    float* dst = partials + ((size_t)batch * CHUNKS + chunk) * 14;
#pragma unroll
    for (int t = 0; t < 14; ++t) dst[t] = red[t][0];
  }
}

// ---------------------------------------------------------------------------
// Kernel 2: per-batch finalize. Reduce CHUNKS partials, compute mean/std,
// gram, top eigenvector (power iteration, PSD matrix), sign fix, and emit
// fused affine coefficients {w0..w3, bias}.
// ---------------------------------------------------------------------------
__global__ __launch_bounds__(CHUNKS) void bpca_finalize(const float* __restrict__ partials,
                                                        float* __restrict__ params) {
  const int batch = blockIdx.x;
  const int tid = threadIdx.x;   // CHUNKS = 128 threads

  __shared__ float red[14][CHUNKS];
  {
    const float* src = partials + ((size_t)batch * CHUNKS + tid) * 14;
#pragma unroll
    for (int t = 0; t < 14; ++t) red[t][tid] = src[t];
  }
  __syncthreads();
  for (int off = CHUNKS / 2; off > 0; off >>= 1) {
    if (tid < off) {
#pragma unroll
      for (int t = 0; t < 14; ++t) red[t][tid] += red[t][tid + off];
    }
    __syncthreads();
  }

  if (tid == 0) {
    const float N = (float)NROWS;
    float S[4], Sg[10];
#pragma unroll
    for (int t = 0; t < 4; ++t) S[t] = red[t][0];
#pragma unroll
    for (int t = 0; t < 10; ++t) Sg[t] = red[4 + t][0];

    const int diag[4] = {0, 4, 7, 9};
    float mu[4], inv[4];
#pragma unroll
    for (int j = 0; j < 4; ++j) {
      mu[j] = S[j] / N;
      float var = Sg[diag[j]] / N - mu[j] * mu[j];
      var = var > 0.0f ? var : 0.0f;
      float sd = sqrtf(var);
      inv[j] = (sd == 0.0f) ? 0.0f : 1.0f / sd;
    }

    // Gram of normalized columns
    float G[4][4];
    int t = 0;
    for (int i = 0; i < 4; ++i)
      for (int j = i; j < 4; ++j, ++t) {
        float g = (Sg[t] - N * mu[i] * mu[j]) * inv[i] * inv[j];
        G[i][j] = g;
        G[j][i] = g;
      }

    // Power iteration for the dominant eigenvector (gram is PSD).
    float v[4] = {0.5f, 0.5f, 0.5f, 0.5f};
    for (int it = 0; it < 256; ++it) {
      float u[4];
#pragma unroll
      for (int i = 0; i < 4; ++i)
        u[i] = G[i][0] * v[0] + G[i][1] * v[1] + G[i][2] * v[2] + G[i][3] * v[3];
      float n2 = u[0] * u[0] + u[1] * u[1] + u[2] * u[2] + u[3] * u[3];
      if (n2 > 0.0f) {
        float rn = 1.0f / sqrtf(n2);
#pragma unroll
        for (int i = 0; i < 4; ++i) v[i] = u[i] * rn;
      }
    }
    float sgn = (v[0] + v[1] + v[2] + v[3]) < 0.0f ? -1.0f : 1.0f;

    float w[4], bias = 0.0f;
#pragma unroll
    for (int j = 0; j < 4; ++j) {
      w[j] = sgn * v[j] * inv[j];
      bias -= w[j] * mu[j];
    }

    float* dst = params + batch * 8;
    dst[0] = w[0]; dst[1] = w[1]; dst[2] = w[2]; dst[3] = w[3]; dst[4] = bias;
  }
}

// ---------------------------------------------------------------------------
// Kernel 3: apply. out[16 rows] = A(16x4) * w(4) (+ bias at store) per group
// via V_WMMA_F32_16X16X4_F32 with C = inline 0.
//   * batch on blockIdx.y -> params become scalar (s_load), hoisted.
//   * 4 consecutive groups (= the four (p1,p2) pixels of one pooled block)
//     per wave-iteration: two contiguous 512B input segments, 256B output,
//     4-deep load clause + 4 independent WMMAs for latency hiding.
//   * A operand loaded directly in the documented 16x4 f32 layout
//     (lane m -> K0/K1, lane m+16 -> K2/K3): coalesced global_load_b64.
// ---------------------------------------------------------------------------
#define APPLY_BLK 256                 // 8 waves per block
#define APPLY_BPB 64                  // blocks per batch -> 512 waves per batch
#define G4_PER_BATCH (NROWS / 64)     // 4096 groups-of-4 per batch
#define APPLY_ITERS (G4_PER_BATCH / (APPLY_BPB * (APPLY_BLK / 32)))   // 8

__global__ __launch_bounds__(APPLY_BLK) void bpca_apply(const float* __restrict__ in,
                                                        const float* __restrict__ params,
                                                        float* __restrict__ out) {
  const int b = blockIdx.y;
  const int lane = threadIdx.x & 31;
  const int m = lane & 15;
  const int hi = lane >> 4;
  const int wid = blockIdx.x * (APPLY_BLK / 32) + (threadIdx.x >> 5);  // 0..511
  const int nw = APPLY_BPB * (APPLY_BLK / 32);                        // 512

  const float* pp = params + b * 8;   // uniform -> scalar loads, hoisted
  const float w0 = pp[0], w1 = pp[1], w2 = pp[2], w3 = pp[3], bias = pp[4];

  const float* inb = in + (size_t)b * IN_PER_B;
  float* outb = out + (size_t)b * OUT_PER_B;

#ifdef HAS_WMMA_F32_4
  v2f bv;
  bv.x = hi ? w2 : w0;
  bv.y = hi ? w3 : w1;
  const v8f cz = {};   // C = 0 (inline constant SRC2)
#endif

#pragma unroll
  for (int it = 0; it < APPLY_ITERS; ++it) {
    const int g4 = wid + it * nw;            // 0..4095: one 2x2 pooled block
    const int wcol = g4 & 63;
    const int hrow = g4 >> 6;
    const float* r00 = inb + ((size_t)(2 * hrow) * PW + 2 * wcol) * PC;
    // group j = (p1<<1)|p2: offsets 0, +PC, +PW*PC, +PW*PC+PC
#ifdef HAS_WMMA_F32_4
    const int aoff = m * 4 + hi * 2;
    v2f a0 = *(const v2f*)(r00 + aoff);
    v2f a1 = *(const v2f*)(r00 + PC + aoff);
    v2f a2 = *(const v2f*)(r00 + PW * PC + aoff);
    v2f a3 = *(const v2f*)(r00 + PW * PC + PC + aoff);
    v8f d0 = __builtin_amdgcn_wmma_f32_16x16x4_f32(false, a0, false, bv, (short)0, cz, false, false);
    v8f d1 = __builtin_amdgcn_wmma_f32_16x16x4_f32(false, a1, false, bv, (short)0, cz, false, false);
    v8f d2 = __builtin_amdgcn_wmma_f32_16x16x4_f32(false, a2, false, bv, (short)0, cz, false, false);
    v8f d3 = __builtin_amdgcn_wmma_f32_16x16x4_f32(false, a3, false, bv, (short)0, cz, false, false);
    if (m == 0) {
      // lane 0: out rows 0..7 of each group; lane 16: rows 8..15
      float* ob = outb + (size_t)g4 * 64 + hi * 8;
      *(float4*)(ob + 0)      = make_float4(d0[0] + bias, d0[1] + bias, d0[2] + bias, d0[3] + bias);
      *(float4*)(ob + 4)      = make_float4(d0[4] + bias, d0[5] + bias, d0[6] + bias, d0[7] + bias);
      *(float4*)(ob + 16)     = make_float4(d1[0] + bias, d1[1] + bias, d1[2] + bias, d1[3] + bias);
      *(float4*)(ob + 20)     = make_float4(d1[4] + bias, d1[5] + bias, d1[6] + bias, d1[7] + bias);
      *(float4*)(ob + 32)     = make_float4(d2[0] + bias, d2[1] + bias, d2[2] + bias, d2[3] + bias);
      *(float4*)(ob + 36)     = make_float4(d2[4] + bias, d2[5] + bias, d2[6] + bias, d2[7] + bias);
      *(float4*)(ob + 48)     = make_float4(d3[0] + bias, d3[1] + bias, d3[2] + bias, d3[3] + bias);
      *(float4*)(ob + 52)     = make_float4(d3[4] + bias, d3[5] + bias, d3[6] + bias, d3[7] + bias);
    }
#else
    if (lane < 16) {
      float* ob = outb + (size_t)g4 * 64;
#pragma unroll
      for (int j = 0; j < 4; ++j) {
        const float* rb = r00 + (j & 1) * PC + ((j >> 1) & 1) * PW * PC;
        float4 p = *(const float4*)(rb + m * 4);
        ob[j * 16 + m] = fmaf(p.x, w0, fmaf(p.y, w1, fmaf(p.z, w2, fmaf(p.w, w3, bias))));
      }
    }
#endif
  }
}

// ---------------------------------------------------------------------------
extern "C" void kernel_launch(void* const* d_in, const int* in_sizes, int n_in,
                              void* d_out, int out_size, void* d_ws, size_t ws_size,
                              hipStream_t stream) {
  (void)in_sizes; (void)n_in; (void)out_size; (void)ws_size;
  const float* in = (const float*)d_in[0];
  float* out = (float*)d_out;

  float* partials = (float*)d_ws;                               // 32*128*14 floats
  float* params = partials + (size_t)NBATCH * CHUNKS * 14;      // 32*8 floats

  bpca_reduce<<<dim3(CHUNKS, NBATCH), BLKA, 0, stream>>>(in, partials);
  bpca_finalize<<<NBATCH, CHUNKS, 0, stream>>>(partials, params);
  bpca_apply<<<dim3(APPLY_BPB, NBATCH), APPLY_BLK, 0, stream>>>(in, params, out);
}